// AllAtomFAPE_49520972922935
// MI455X (gfx1250) — compile-verified
//
#include <hip/hip_runtime.h>
#include <hip/hip_bf16.h>
#include <math.h>

typedef __attribute__((ext_vector_type(2))) float v2f;
typedef __attribute__((ext_vector_type(8))) float v8f;

#define NB    2
#define NRES  512
#define NATOM 7168            // 512 * 14
#define KDIM  8               // K padded to 8 (two 16x16x4 WMMA steps)
#define D_CLAMP 10.0f
#define FAPE_EPS 1e-4f
#define ZNORM 10.0f

// Workspace layout (floats):
//  [0..1]                      per-batch accumulator
//  [16 .. 16+NB*3*NRES*KDIM)   A_all[b][o][i][k]   (24576 floats)
//  [.. + NB*KDIM*NATOM)        Bmat [b][k][j]      (114688 floats)
#define WS_ACC 0
#define WS_A   16
#define WS_B   (WS_A + NB * 3 * NRES * KDIM)

// ---------------------------------------------------------------------------
// Prep: build WMMA operand matrices, zero accumulators.
// ---------------------------------------------------------------------------
__global__ void fape_prep(const float* __restrict__ pred_rot,
                          const float* __restrict__ pred_trans,
                          const float* __restrict__ pred_pos,
                          const float* __restrict__ true_rot,
                          const float* __restrict__ true_trans,
                          const float* __restrict__ true_pos,
                          float* __restrict__ ws) {
  const int tid = blockIdx.x * blockDim.x + threadIdx.x;
  if (tid < NB) ws[WS_ACC + tid] = 0.0f;

  const int nA = NB * 3 * NRES;          // 3072 A-matrix rows
  if (tid < nA) {
    const int b   = tid / (3 * NRES);
    const int rem = tid - b * 3 * NRES;
    const int o   = rem / NRES;
    const int i   = rem - o * NRES;
    const float* Rp = pred_rot   + (size_t)(b * NRES + i) * 9;
    const float* Rt = true_rot   + (size_t)(b * NRES + i) * 9;
    const float* tp = pred_trans + (size_t)(b * NRES + i) * 3;
    const float* tt = true_trans + (size_t)(b * NRES + i) * 3;
    float* A = ws + WS_A + ((size_t)(b * 3 + o) * NRES + i) * KDIM;
    float cst = 0.0f;
#pragma unroll
    for (int c = 0; c < 3; ++c) {
      const float rp = Rp[c * 3 + o];    // R_pred_inv[o][c] = R_pred[c][o]
      const float rt = Rt[c * 3 + o];
      A[c]     =  rp;
      A[3 + c] = -rt;
      cst += -rp * tp[c] + rt * tt[c];   // t_pred_inv[o] - t_true_inv[o]
    }
    A[6] = cst;
    A[7] = 0.0f;
  } else if (tid < nA + NB * NATOM) {
    const int t = tid - nA;
    const int b = t / NATOM;
    const int j = t - b * NATOM;
    const float* pp = pred_pos + (size_t)(b * NATOM + j) * 3;
    const float* qq = true_pos + (size_t)(b * NATOM + j) * 3;
    float* B = ws + WS_B + (size_t)b * KDIM * NATOM;
    B[0 * NATOM + j] = pp[0];
    B[1 * NATOM + j] = pp[1];
    B[2 * NATOM + j] = pp[2];
    B[3 * NATOM + j] = qq[0];
    B[4 * NATOM + j] = qq[1];
    B[5 * NATOM + j] = qq[2];
    B[6 * NATOM + j] = 1.0f;
    B[7 * NATOM + j] = 0.0f;
  }
}

// ---------------------------------------------------------------------------
// Main: one wave per 16(frames)x16(atoms) tile.
// 3 output dims x 2 K-steps = 6 x V_WMMA_F32_16X16X4_F32 per tile.
// Grid: (NATOM/16, NRES/16/8, NB), block 256 (8 waves).
// ---------------------------------------------------------------------------
__global__ void __launch_bounds__(256, 4)
fape_main(const float* __restrict__ mask,
          const float* __restrict__ ws,
          float* __restrict__ acc) {
  const int lane = threadIdx.x & 31;
  const int wave = threadIdx.x >> 5;
  const int it   = blockIdx.y * 8 + wave;   // frame tile 0..31
  const int jt   = blockIdx.x;              // atom tile  0..447
  const int b    = blockIdx.z;

  const int l15 = lane & 15;
  const int row = it * 16 + l15;            // frame index for A fetch
  const int col = jt * 16 + l15;            // atom index (C column = lane&15)
  const int hk  = (lane < 16) ? 0 : 2;      // ISA A/B layout: v0 holds K{0,2}, v1 K{1,3}

  const float* Bb = ws + WS_B + (size_t)b * KDIM * NATOM;
  const float* A0 = ws + WS_A + ((size_t)(b * 3 + 0) * NRES + row) * KDIM;
  const float* A1 = ws + WS_A + ((size_t)(b * 3 + 1) * NRES + row) * KDIM;
  const float* A2 = ws + WS_A + ((size_t)(b * 3 + 2) * NRES + row) * KDIM;

  v8f c0 = {}, c1 = {}, c2 = {};
#pragma unroll
  for (int s = 0; s < 2; ++s) {            // two K=4 steps cover K=8
    const int k0 = 4 * s + hk;
    v2f bm;
    bm.x = Bb[(size_t)(k0    ) * NATOM + col];
    bm.y = Bb[(size_t)(k0 + 1) * NATOM + col];
    v2f a;
    a.x = A0[k0]; a.y = A0[k0 + 1];
    c0 = __builtin_amdgcn_wmma_f32_16x16x4_f32(false, a, false, bm, (short)0, c0, false, false);
    a.x = A1[k0]; a.y = A1[k0 + 1];
    c1 = __builtin_amdgcn_wmma_f32_16x16x4_f32(false, a, false, bm, (short)0, c1, false, false);
    a.x = A2[k0]; a.y = A2[k0 + 1];
    c2 = __builtin_amdgcn_wmma_f32_16x16x4_f32(false, a, false, bm, (short)0, c2, false, false);
  }

  // Epilogue: dist = min(sqrt(dx^2+dy^2+dz^2 + eps), 10) * mask[col].
  // c0/c1/c2 share the (frame, atom) layout, so this is purely elementwise.
  const float msk = mask[(size_t)b * NATOM + col];
  float part = 0.0f;
#pragma unroll
  for (int e = 0; e < 8; ++e) {
    const float d2 = c0[e] * c0[e] + c1[e] * c1[e] + c2[e] * c2[e];
    const float dist = sqrtf(d2 + FAPE_EPS);
    part += fminf(dist, D_CLAMP) * msk;
  }

  // wave32 reduction, then one atomic per block.
#pragma unroll
  for (int off = 16; off > 0; off >>= 1)
    part += __shfl_xor(part, off, 32);

  __shared__ float sp[8];
  if (lane == 0) sp[wave] = part;
  __syncthreads();
  if (threadIdx.x == 0) {
    float s = 0.0f;
#pragma unroll
    for (int w = 0; w < 8; ++w) s += sp[w];
    atomicAdd(&acc[b], s);
  }
}

// ---------------------------------------------------------------------------
// Finalize: out[b] = acc[b] / max(sum(mask_b),1) / NRES / Z. One wave per batch.
// ---------------------------------------------------------------------------
__global__ void fape_final(const float* __restrict__ mask,
                           const float* __restrict__ acc,
                           float* __restrict__ out) {
  const int b = blockIdx.x;
  const int lane = threadIdx.x;
  float s = 0.0f;
  for (int j = lane; j < NATOM; j += 32) s += mask[(size_t)b * NATOM + j];
#pragma unroll
  for (int off = 16; off > 0; off >>= 1)
    s += __shfl_xor(s, off, 32);
  if (lane == 0) {
    const float n = fmaxf(s, 1.0f);
    out[b] = acc[b] / n / (float)NRES / ZNORM;
  }
}

// ---------------------------------------------------------------------------
extern "C" void kernel_launch(void* const* d_in, const int* in_sizes, int n_in,
                              void* d_out, int out_size, void* d_ws, size_t ws_size,
                              hipStream_t stream) {
  const float* pred_rot   = (const float*)d_in[0];
  const float* pred_trans = (const float*)d_in[1];
  const float* pred_pos   = (const float*)d_in[2];
  const float* atom_mask  = (const float*)d_in[3];
  const float* true_rot   = (const float*)d_in[4];
  const float* true_trans = (const float*)d_in[5];
  const float* true_pos   = (const float*)d_in[6];
  float* out = (float*)d_out;
  float* ws  = (float*)d_ws;

  // prep covers A-rows (3072) + B-columns (14336)
  const int prep_threads = NB * 3 * NRES + NB * NATOM;
  fape_prep<<<(prep_threads + 255) / 256, 256, 0, stream>>>(
      pred_rot, pred_trans, pred_pos, true_rot, true_trans, true_pos, ws);

  dim3 grid(NATOM / 16, (NRES / 16) / 8, NB);   // 448 x 4 x 2
  fape_main<<<grid, 256, 0, stream>>>(atom_mask, ws, ws + WS_ACC);

  fape_final<<<NB, 32, 0, stream>>>(atom_mask, ws + WS_ACC, out);
}